// GlobalRankLoss_53231824667135
// MI455X (gfx1250) — compile-verified
//
#include <hip/hip_runtime.h>

// GlobalRankLoss on MI455X (gfx1250).
// fp32 GEMM 8192x4096x512 reduced on the fly to per-row pos/neg sums using
// V_WMMA_F32_16X16X4_F32 (fp32 WMMA -> numerics match the fp32 reference).
// Each workgroup owns one batch b (32 face rows): two A fragments share every
// B fragment, so each global B load feeds 2 WMMAs (halves L2 traffic vs the
// 16-row version). A strip lives in LDS with a 516-dword row stride so a b64
// fragment load hits all 64 banks exactly once per wave.

typedef __attribute__((ext_vector_type(2))) float v2f;
typedef __attribute__((ext_vector_type(8))) float v8f;

#define B_SZ    256
#define F_SZ    32
#define N_SZ    16
#define D_SZ    512
#define A_PITCH 516                     // padded row stride (dwords), bank-conflict free
#define NTILES  (B_SZ)                  // 256 column tiles of 16 (one per batch c)

__global__ __launch_bounds__(256) void grl_main(const float* __restrict__ face,
                                                const float* __restrict__ ner,
                                                float* __restrict__ blk_out)
{
    __shared__ float As[32 * A_PITCH];  // ~64.5 KB: 32 rows (one batch) x 512 (+4 pad)
    __shared__ float maskS[32];
    __shared__ float negW[8][32];
    __shared__ float posW[8][32];
    __shared__ float rankS[32];

    const int tid  = threadIdx.x;
    const int wave = tid >> 5;
    const int lane = tid & 31;
    const int b    = blockIdx.x;        // batch: rows b*32 .. b*32+31

    // ---- Stage A (32 x 512) into padded LDS ----
    {
        const float4* gA = reinterpret_cast<const float4*>(face + (size_t)b * 32 * D_SZ);
        for (int i = tid; i < 32 * D_SZ / 4; i += 256) {
            float4 v = gA[i];
            const int row = i / (D_SZ / 4);
            const int col = (i % (D_SZ / 4)) * 4;
            float* dst = &As[row * A_PITCH + col];
            dst[0] = v.x; dst[1] = v.y; dst[2] = v.z; dst[3] = v.w;
        }
    }
    __syncthreads();

    // ---- face_mask: 8 waves x 2 passes x 2 rows (16-lane groups) ----
    #pragma unroll
    for (int pass = 0; pass < 2; ++pass) {
        const int row = wave * 4 + pass * 2 + (lane >> 4);
        const int l16 = lane & 15;
        const float* rp = &As[row * A_PITCH];
        float sum = 0.f;
        for (int i = l16; i < D_SZ; i += 16) sum += rp[i];
        #pragma unroll
        for (int m = 8; m >= 1; m >>= 1) sum += __shfl_xor(sum, m, 16);
        if (l16 == 0) maskS[row] = (sum != 0.0f) ? 1.0f : 0.0f;
    }

    // ---- Per-lane row accumulators for both 16-row halves ----
    float accNeg0[8], accPos0[8], accNeg1[8], accPos1[8];
    #pragma unroll
    for (int i = 0; i < 8; ++i) {
        accNeg0[i] = 0.f; accPos0[i] = 0.f;
        accNeg1[i] = 0.f; accPos1[i] = 0.f;
    }

    // A fragment (16x4 fp32): lanes 0-15 hold K=k0..k0+1 of row lane&15,
    // lanes 16-31 hold K=k0+2..k0+3 of the same row. Two fragments: rows 0-15 / 16-31.
    const int rowA = lane & 15;
    const int koff = (lane >> 4) << 1;
    const float* aRow0 = &As[rowA * A_PITCH];
    const float* aRow1 = &As[(rowA + 16) * A_PITCH];
    const int nB = lane & 15;           // B fragment column

    for (int t = wave; t < NTILES; t += 8) {        // wave-uniform control
        const float* bBase = ner + ((size_t)t * 16 + nB) * D_SZ;
        if (t + 8 < NTILES)                          // prefetch next B tile row
            __builtin_prefetch(ner + ((size_t)(t + 8) * 16 + nB) * D_SZ, 0, 3);

        v8f acc0 = {0.f,0.f,0.f,0.f,0.f,0.f,0.f,0.f};
        v8f acc1 = {0.f,0.f,0.f,0.f,0.f,0.f,0.f,0.f};
        #pragma unroll 4
        for (int k0 = 0; k0 < D_SZ; k0 += 4) {
            const int d = k0 + koff;
            v2f bb, a0, a1;
            bb.x = bBase[d];  bb.y = bBase[d + 1];   // 1 global_load_b64 ...
            a0.x = aRow0[d];  a0.y = aRow0[d + 1];   // ds loads, 64-bank clean
            a1.x = aRow1[d];  a1.y = aRow1[d + 1];
            acc0 = __builtin_amdgcn_wmma_f32_16x16x4_f32(
                       false, a0, false, bb, (short)0, acc0, false, false);
            acc1 = __builtin_amdgcn_wmma_f32_16x16x4_f32(   // ... feeds 2 WMMAs
                       false, a1, false, bb, (short)0, acc1, false, false);
        }
        if (t == b) {                               // diagonal tile -> pos_sim
            #pragma unroll
            for (int v = 0; v < 8; ++v) {
                accPos0[v] += fmaxf(acc0[v], 0.f);
                accPos1[v] += fmaxf(acc1[v], 0.f);
            }
        } else {                                    // off-diagonal -> neg sum
            #pragma unroll
            for (int v = 0; v < 8; ++v) {
                float m0 = fmaxf(acc0[v], 0.f);
                float m1 = fmaxf(acc1[v], 0.f);
                m0 = (m0 == 1.0f) ? 0.f : m0;       // sims_m masking
                m1 = (m1 == 1.0f) ? 0.f : m1;
                accNeg0[v] += m0;
                accNeg1[v] += m1;
            }
        }
    }

    // ---- Reduce across the 16 lanes of each half (tile columns) ----
    #pragma unroll
    for (int v = 0; v < 8; ++v) {
        float n0 = accNeg0[v], p0 = accPos0[v], n1 = accNeg1[v], p1 = accPos1[v];
        #pragma unroll
        for (int m = 8; m >= 1; m >>= 1) {
            n0 += __shfl_xor(n0, m, 16); p0 += __shfl_xor(p0, m, 16);
            n1 += __shfl_xor(n1, m, 16); p1 += __shfl_xor(p1, m, 16);
        }
        accNeg0[v] = n0; accPos0[v] = p0; accNeg1[v] = n1; accPos1[v] = p1;
    }
    if ((lane & 15) == 0) {
        const int rbase = (lane >> 4) * 8;   // lane 0 -> rows 0-7, lane 16 -> rows 8-15
        #pragma unroll
        for (int v = 0; v < 8; ++v) {
            negW[wave][rbase + v]      = accNeg0[v];
            posW[wave][rbase + v]      = accPos0[v];
            negW[wave][16 + rbase + v] = accNeg1[v];
            posW[wave][16 + rbase + v] = accPos1[v];
        }
    }
    __syncthreads();

    // ---- Combine waves in fixed order (deterministic), compute rank ----
    if (tid < 32) {
        float n = 0.f, p = 0.f;
        #pragma unroll
        for (int w = 0; w < 8; ++w) { n += negW[w][tid]; p += posW[w][tid]; }
        const float neg_sim = n * (1.0f / (255.0f + 1e-8f));
        rankS[tid] = fmaxf(neg_sim - p + 0.2f, 0.f) * maskS[tid];
    }
    __syncthreads();
    if (tid == 0) {
        float sum = 0.f;
        #pragma unroll
        for (int i = 0; i < 32; ++i) sum += rankS[i];
        blk_out[b] = sum;
    }
}

__global__ __launch_bounds__(256) void grl_finish(const float* __restrict__ blk_in,
                                                  float* __restrict__ out)
{
    __shared__ float red[256];
    const int tid = threadIdx.x;
    red[tid] = blk_in[tid];             // 256 per-batch partials
    __syncthreads();
    for (int ofs = 128; ofs >= 1; ofs >>= 1) {
        if (tid < ofs) red[tid] += red[tid + ofs];
        __syncthreads();
    }
    if (tid == 0) out[0] = red[0] * (1.0f / 256.0f);   // BETA * sum / B
}

extern "C" void kernel_launch(void* const* d_in, const int* in_sizes, int n_in,
                              void* d_out, int out_size, void* d_ws, size_t ws_size,
                              hipStream_t stream)
{
    const float* face = (const float*)d_in[0];   // (256, 32, 512) f32
    const float* ner  = (const float*)d_in[1];   // (256, 16, 512) f32
    float* blk = (float*)d_ws;                   // 256 floats of scratch

    grl_main<<<B_SZ, 256, 0, stream>>>(face, ner, blk);
    grl_finish<<<1, 256, 0, stream>>>(blk, (float*)d_out);
}